// AttentionLayer_77970836292244
// MI455X (gfx1250) — compile-verified
//
#include <hip/hip_runtime.h>
#include <hip/hip_bf16.h>

typedef __attribute__((ext_vector_type(2))) float v2f;
typedef __attribute__((ext_vector_type(8))) float v8f;

#define BATCH 4
#define TQ 512
#define TK 512
#define DIM 128
#define QT 16            // query rows per workgroup (WMMA M)
#define KT 128           // k-tile staged in LDS
#define VS_STRIDE 129    // Vs row stride (transposed V tile), conflict-free reads
#define ES_STRIDE 516    // Es row stride, bank-stride 4 for column reads
#define NTHREADS 256

// ---- native tanh: gfx1250 has V_TANH_F32 (trans op). Fallback uses the
// identity tanh(x) = 1 - 2/(exp(2x)+1) with native v_exp_f32 / v_rcp_f32
// (~4 ops), never the ~25-op OCML polynomial. ----
#if __has_builtin(__builtin_amdgcn_tanhf)
__device__ __forceinline__ float fast_tanh(float x) { return __builtin_amdgcn_tanhf(x); }
#elif __has_builtin(__builtin_amdgcn_tanh_f32)
__device__ __forceinline__ float fast_tanh(float x) { return __builtin_amdgcn_tanh_f32(x); }
#else
__device__ __forceinline__ float fast_tanh(float x) {
    // exp(2x)->inf  => 1 - 0  = +1 ; exp(2x)->0 => 1 - 2 = -1  (correct saturation)
    return 1.0f - 2.0f / (__expf(2.0f * x) + 1.0f);
}
#endif

// ---- CDNA5 async VGLOBAL -> LDS copies (ASYNCcnt-tracked, no VGPR roundtrip) ----
// LDS operand is the low 32 bits of the flat address (ISA 10.2: LDS aperture
// uses addr[31:0]); "off" = GV mode (full 64-bit VGPR address).
__device__ __forceinline__ void async_ld_lds_b32(float* lds_dst, const float* gsrc) {
    unsigned lo = (unsigned)(uintptr_t)lds_dst;
    asm volatile("global_load_async_to_lds_b32 %0, %1, off"
                 :: "v"(lo), "v"(gsrc) : "memory");
}
__device__ __forceinline__ void async_ld_lds_b128(float* lds_dst, const float* gsrc) {
    unsigned lo = (unsigned)(uintptr_t)lds_dst;
    asm volatile("global_load_async_to_lds_b128 %0, %1, off"
                 :: "v"(lo), "v"(gsrc) : "memory");
}
__device__ __forceinline__ void wait_async0() {
    asm volatile("s_wait_asynccnt 0" ::: "memory");
}

// Stage one 128-row V tile into LDS, transposed: Vs[d][k] = V[kbase+k][d].
// Global side: lanes walk consecutive d within a k row (coalesced 128B/wave);
// LDS side: 4B scatter into the transposed layout, done by the async engine.
__device__ __forceinline__ void stage_v_tile_async(const float* __restrict__ Vb,
                                                   float* __restrict__ Vs,
                                                   int kt, int t) {
    const float* src = Vb + (size_t)kt * KT * DIM;
    #pragma unroll 4
    for (int i = t; i < KT * DIM; i += NTHREADS) {
        int kk = i >> 7;             // row in tile
        int d  = i & 127;            // lane-consecutive -> coalesced
        async_ld_lds_b32(Vs + d * VS_STRIDE + kk, src + kk * DIM + d);
    }
}

__global__ __launch_bounds__(NTHREADS)
void addattn_kernel(const float* __restrict__ q,
                    const float* __restrict__ v,
                    float* __restrict__ out) {
    __shared__ float Qs[QT * DIM];            //  8 KB
    __shared__ float Vs[DIM * VS_STRIDE];     // 66 KB (transposed V tile)
    __shared__ float Es[QT * ES_STRIDE];      // 33 KB (scores -> weights in place)

    const int t     = threadIdx.x;
    const int b     = blockIdx.x >> 5;        // 32 q-tiles per batch
    const int qtile = blockIdx.x & 31;
    const int qbase = qtile * QT;

    const float* Vb = v + (size_t)b * TK * DIM;
    const float* Qb = q + ((size_t)b * TQ + qbase) * DIM;

    // ---- async-load Q tile (contiguous 16B chunks) ----
    for (int i = t; i < QT * (DIM / 4); i += NTHREADS) {
        int qq = i >> 5;
        int d4 = (i & 31) * 4;
        async_ld_lds_b128(Qs + qq * DIM + d4, Qb + qq * DIM + d4);
    }

    // ==== Phase B: scores e[q,k] = sum_d tanh(Q[q,d] + V[k,d]) ====
    for (int kt = 0; kt < TK / KT; ++kt) {
        __syncthreads();                       // Vs safe to overwrite
        stage_v_tile_async(Vb, Vs, kt, t);
        wait_async0();                         // all copies (incl. Q) landed
        __syncthreads();
        #pragma unroll
        for (int s = 0; s < (QT * KT) / NTHREADS; ++s) {
            int p  = s * NTHREADS + t;
            int qq = p >> 7;                   // constant per wave
            int kk = p & 127;                  // lane-consecutive
            const float* qrow = Qs + qq * DIM; // broadcast reads
            const float* vcol = Vs + kk;       // conflict-free reads
            float acc = 0.f;
            #pragma unroll 4
            for (int d = 0; d < DIM; ++d)
                acc += fast_tanh(qrow[d] + vcol[d * VS_STRIDE]);
            Es[qq * ES_STRIDE + kt * KT + kk] = acc;
        }
    }
    __syncthreads();

    // ==== Phase C: row softmax over k, normalize Es in place ====
    {
        int wave = t >> 5, lane = t & 31;
        for (int r = 0; r < 2; ++r) {          // 8 waves x 2 rows = 16 rows
            int qq = wave * 2 + r;
            float* erow = Es + qq * ES_STRIDE;
            float e[16];
            float m = -3.4e38f;
            #pragma unroll
            for (int j = 0; j < 16; ++j) { e[j] = erow[lane + 32 * j]; m = fmaxf(m, e[j]); }
            #pragma unroll
            for (int off = 16; off > 0; off >>= 1) m = fmaxf(m, __shfl_xor(m, off, 32));
            float ssum = 0.f;
            #pragma unroll
            for (int j = 0; j < 16; ++j) { e[j] = __expf(e[j] - m); ssum += e[j]; }
            #pragma unroll
            for (int off = 16; off > 0; off >>= 1) ssum += __shfl_xor(ssum, off, 32);
            float inv = 1.0f / ssum;
            #pragma unroll
            for (int j = 0; j < 16; ++j) erow[lane + 32 * j] = e[j] * inv;
        }
    }
    __syncthreads();

    // ==== Phase D: O[16,128] = W[16,512] @ V[512,128] via fp32 WMMA ====
    const int wave  = t >> 5;
    const int lane  = t & 31;
    const int m16   = lane & 15;           // A row / B col within 16
    const int khalf = (lane >> 4) * 2;     // K pair owned by this half-wave
    const int d0    = wave * 16;           // each wave owns one 16-wide d tile
    v8f cacc = {};                         // fp32 accumulator (8 VGPRs)

    for (int kt = 0; kt < TK / KT; ++kt) {
        __syncthreads();                   // previous Vs readers done
        stage_v_tile_async(Vb, Vs, kt, t);
        wait_async0();
        __syncthreads();
        #pragma unroll 4
        for (int k4 = 0; k4 < KT / 4; ++k4) {
            int kk = k4 * 4 + khalf;
            v2f a, bb;
            // A (16x4): W[qbase+m16][kbase+kk..kk+1]  (bank-stride-4 column read)
            a.x = Es[m16 * ES_STRIDE + kt * KT + kk];
            a.y = Es[m16 * ES_STRIDE + kt * KT + kk + 1];
            // B (4x16): V[kbase+kk..kk+1][d0+m16]     (stride-129 => distinct banks)
            bb.x = Vs[(d0 + m16) * VS_STRIDE + kk];
            bb.y = Vs[(d0 + m16) * VS_STRIDE + kk + 1];
            cacc = __builtin_amdgcn_wmma_f32_16x16x4_f32(
                /*neg_a=*/false, a, /*neg_b=*/false, bb,
                /*c_mod=*/(short)0, cacc, /*reuse_a=*/false, /*reuse_b=*/false);
        }
    }

    // ---- store D: VGPR r, lane l -> M = r + 8*(l>=16), N = l%16 ----
    float* orow = out + ((size_t)b * TQ + qbase) * DIM;
    #pragma unroll
    for (int r = 0; r < 8; ++r) {
        int mm = r + 8 * (lane >> 4);
        orow[mm * DIM + d0 + m16] = cacc[r];
    }
}

extern "C" void kernel_launch(void* const* d_in, const int* in_sizes, int n_in,
                              void* d_out, int out_size, void* d_ws, size_t ws_size,
                              hipStream_t stream) {
    (void)in_sizes; (void)n_in; (void)d_ws; (void)ws_size; (void)out_size;
    const float* q = (const float*)d_in[0];
    const float* v = (const float*)d_in[1];
    float* out = (float*)d_out;
    dim3 grid(BATCH * (TQ / QT));   // 128 workgroups
    addattn_kernel<<<grid, NTHREADS, 0, stream>>>(q, v, out);
}